// SparseBox3DLoss_34308198760689
// MI455X (gfx1250) — compile-verified
//
#include <hip/hip_runtime.h>
#include <math.h>

#define MAXV 8
#define EPSF 1e-8f
#define BLOCK 256
#define ROWF 11

// fast reciprocal: v_rcp_f32 (~1 ulp), plenty for interior intersection points
__device__ __forceinline__ float frcp(float x) { return __builtin_amdgcn_rcpf(x); }

// write (vx,vy) at dynamic position idx; chain statically bounded by UB.
// idx == -1 (non-emit) matches nothing; idx >= UB silently drops
// (== reference's at[].set(mode="drop") at position >= 8).
template <int UB>
__device__ __forceinline__ void pset(float* px, float* py, int idx, float vx, float vy) {
#pragma unroll
    for (int k = 0; k < UB; ++k) {
        const bool w = (k == idx);
        px[k] = w ? vx : px[k];
        py[k] = w ? vy : py[k];
    }
}

// ---- clip 0: subject is the raw rect, count == 4 at compile time ----
template <int I>
__device__ __forceinline__ void clip4_step(const float* vx, const float* vy,
                                           float ax, float ay, float dx, float dy,
                                           float& pvx, float& pvy, bool& pin,
                                           float* ox, float* oy, int& oc,
                                           float& lkx, float& lky) {
    if constexpr (I < 4) {
        const float cx = vx[I], cy = vy[I];
        const bool cin = (dx * (cy - ay) - dy * (cx - ax)) >= 0.f;
        const bool crs = (pin != cin);
        const float num = (ax - pvx) * dy - (ay - pvy) * dx;
        const float den = (cx - pvx) * dy - (cy - pvy) * dx + EPSF;
        const float t = num * frcp(den);
        const float ex = pvx + t * (cx - pvx);
        const float ey = pvy + t * (cy - pvy);
        constexpr int UB = (2 * I + 2 < MAXV) ? (2 * I + 2) : MAXV;  // cursor < 2I+2, exact
        pset<UB>(ox, oy, crs ? oc : -1, ex, ey);
        lkx = crs ? ex : lkx;  lky = crs ? ey : lky;
        oc += (int)crs;
        pset<UB>(ox, oy, cin ? oc : -1, cx, cy);
        lkx = cin ? cx : lkx;  lky = cin ? cy : lky;
        oc += (int)cin;
        pvx = cx; pvy = cy; pin = cin;
        clip4_step<I + 1>(vx, vy, ax, ay, dx, dy, pvx, pvy, pin, ox, oy, oc, lkx, lky);
    }
}

__device__ __forceinline__ int clip_rect4(const float* vx, const float* vy,
                                          float ax, float ay, float bx, float by,
                                          float* ox, float* oy,
                                          float& lastx, float& lasty) {
    const float dx = bx - ax, dy = by - ay;
    float pvx = vx[3], pvy = vy[3];                 // prev = poly[count-1], static
    bool pin = (dx * (pvy - ay) - dy * (pvx - ax)) >= 0.f;
    int oc = 0;
    float lkx = 0.f, lky = 0.f;
    clip4_step<0>(vx, vy, ax, ay, dx, dy, pvx, pvy, pin, ox, oy, oc, lkx, lky);
    lastx = lkx; lasty = lky;
    return (oc > MAXV) ? MAXV : oc;                 // count = min(emits, 8)
}

// ---- clips 1..2: general Sutherland-Hodgman, dynamic count <= 8 ----
template <int I>
__device__ __forceinline__ void clip8_step(const float* px, const float* py, int count,
                                           float ax, float ay, float dx, float dy,
                                           float& pvx, float& pvy, bool& pin,
                                           float* ox, float* oy, int& oc,
                                           float& lkx, float& lky) {
    if constexpr (I < MAXV) {
        if (I < count) {            // wave-uniform skip is common for I >= 5
            const float cx = px[I], cy = py[I];
            const bool cin = (dx * (cy - ay) - dy * (cx - ax)) >= 0.f;
            const bool crs = (pin != cin);
            const float num = (ax - pvx) * dy - (ay - pvy) * dx;
            const float den = (cx - pvx) * dy - (cy - pvy) * dx + EPSF;
            const float t = num * frcp(den);
            const float ex = pvx + t * (cx - pvx);
            const float ey = pvy + t * (cy - pvy);
            constexpr int UB = (2 * I + 2 < MAXV) ? (2 * I + 2) : MAXV;
            constexpr bool CAP = (2 * I + 1 >= MAXV);   // cursor can reach 8 only for I>=4
            pset<UB>(ox, oy, crs ? oc : -1, ex, ey);
            {
                const bool kept = CAP ? (crs && (oc < MAXV)) : crs;
                lkx = kept ? ex : lkx;  lky = kept ? ey : lky;
            }
            oc += (int)crs;
            pset<UB>(ox, oy, cin ? oc : -1, cx, cy);
            {
                const bool kept = CAP ? (cin && (oc < MAXV)) : cin;
                lkx = kept ? cx : lkx;  lky = kept ? cy : lky;
            }
            oc += (int)cin;
            pvx = cx; pvy = cy; pin = cin;
        }
        clip8_step<I + 1>(px, py, count, ax, ay, dx, dy, pvx, pvy, pin, ox, oy, oc, lkx, lky);
    }
}

// lastx/lasty in: prev vertex (== poly[count-1]); out: last kept emit
__device__ __forceinline__ int clip_poly8(const float* px, const float* py, int count,
                                          float ax, float ay, float bx, float by,
                                          float* ox, float* oy,
                                          float& lastx, float& lasty) {
    const float dx = bx - ax, dy = by - ay;
    float pvx = lastx, pvy = lasty;
    bool pin = (dx * (pvy - ay) - dy * (pvx - ax)) >= 0.f;
    int oc = 0;
    float lkx = 0.f, lky = 0.f;
    clip8_step<0>(px, py, count, ax, ay, dx, dy, pvx, pvy, pin, ox, oy, oc, lkx, lky);
    lastx = lkx; lasty = lky;
    return (oc > MAXV) ? MAXV : oc;
}

// ---- clip 3 fused with shoelace: accumulate cross(last,new) in emit order,
//      closing term last -> identical summation order to the reference ----
template <int I>
__device__ __forceinline__ void area_step(const float* px, const float* py, int count,
                                          float ax, float ay, float dx, float dy,
                                          float& pvx, float& pvy, bool& pin,
                                          float& fx, float& fy, float& lx, float& ly,
                                          bool& have, float& acc, int& oc) {
    if constexpr (I < MAXV) {
        if (I < count) {
            const float cx = px[I], cy = py[I];
            const bool cin = (dx * (cy - ay) - dy * (cx - ax)) >= 0.f;
            const bool crs = (pin != cin);
            const float num = (ax - pvx) * dy - (ay - pvy) * dx;
            const float den = (cx - pvx) * dy - (cy - pvy) * dx + EPSF;
            const float t = num * frcp(den);
            const float ex = pvx + t * (cx - pvx);
            const float ey = pvy + t * (cy - pvy);
            constexpr bool CAP = (2 * I + 1 >= MAXV);   // drop possible only for I>=4
            const bool e1 = CAP ? (crs && (oc < MAXV)) : crs;
            acc += (e1 && have) ? (lx * ey - ly * ex) : 0.f;
            fx = (e1 && !have) ? ex : fx;
            fy = (e1 && !have) ? ey : fy;
            lx = e1 ? ex : lx;  ly = e1 ? ey : ly;
            have = have || e1;
            oc += (int)crs;
            const bool e2 = CAP ? (cin && (oc < MAXV)) : cin;
            acc += (e2 && have) ? (lx * cy - ly * cx) : 0.f;
            fx = (e2 && !have) ? cx : fx;
            fy = (e2 && !have) ? cy : fy;
            lx = e2 ? cx : lx;  ly = e2 ? cy : ly;
            have = have || e2;
            oc += (int)cin;
            pvx = cx; pvy = cy; pin = cin;
        }
        area_step<I + 1>(px, py, count, ax, ay, dx, dy, pvx, pvy, pin,
                         fx, fy, lx, ly, have, acc, oc);
    }
}

// returns 2 * signed area of final clipped polygon (exactly 0 whenever kept
// emits < 3, matching the reference's count<3 -> 0 rule)
__device__ __forceinline__ float clip_area_final(const float* px, const float* py, int count,
                                                 float ax, float ay, float bx, float by,
                                                 float lastx, float lasty) {
    const float dx = bx - ax, dy = by - ay;
    float pvx = lastx, pvy = lasty;
    bool pin = (dx * (pvy - ay) - dy * (pvx - ax)) >= 0.f;
    float fx = 0.f, fy = 0.f, lx = 0.f, ly = 0.f;
    bool have = false;
    float acc = 0.f;
    int oc = 0;
    area_step<0>(px, py, count, ax, ay, dx, dy, pvx, pvy, pin,
                 fx, fy, lx, ly, have, acc, oc);
    acc += have ? (lx * fy - ly * fx) : 0.f;          // closing edge, last term
    return acc;
}

// corners (-dx,-dy),(dx,-dy),(dx,dy),(-dx,dy) rotated by (s,c), translated
__device__ __forceinline__ void rect_verts(float cx, float cy, float w, float l,
                                           float s, float c, float* vx, float* vy) {
    const float dx = 0.5f * w, dy = 0.5f * l;
    vx[0] = -dx * c + dy * s + cx;  vy[0] = -dx * s - dy * c + cy;
    vx[1] =  dx * c + dy * s + cx;  vy[1] =  dx * s - dy * c + cy;
    vx[2] =  dx * c - dy * s + cx;  vy[2] =  dx * s + dy * c + cy;
    vx[3] = -dx * c - dy * s + cx;  vy[3] = -dx * s + dy * c + cy;
}

// wave32 shuffle reduction: lane-permute hardware, no LDS memory traffic
__device__ __forceinline__ float wave_reduce_add(float v) {
#pragma unroll
    for (int off = 16; off > 0; off >>= 1)
        v += __shfl_down(v, off, 32);
    return v;
}

__global__ __launch_bounds__(BLOCK)
void SparseBox3DLoss_main_kernel(const float* __restrict__ box,
                                 const float* __restrict__ tgt,
                                 const float* __restrict__ wgt,
                                 float* __restrict__ partial,
                                 int n) {
    __shared__ __align__(16) float sB[BLOCK * ROWF];
    __shared__ __align__(16) float sT[BLOCK * ROWF];
    __shared__ float redA[BLOCK / 32];
    __shared__ float redB[BLOCK / 32];

    const int tid = threadIdx.x;
    const int total = n * ROWF;                        // in_sizes[0] is int -> fits
    const int base  = blockIdx.x * (BLOCK * ROWF);
    const int m     = min(BLOCK * ROWF, total - base); // uniform; multiple of ROWF

    // gfx1250 global_prefetch_b8: warm the next tile
    {
        const int next = base + BLOCK * ROWF + tid * 32;
        if ((tid * 32 < BLOCK * ROWF) && (next < total)) {
            __builtin_prefetch(box + next, 0, 1);
            __builtin_prefetch(tgt + next, 0, 1);
            __builtin_prefetch(wgt + next, 0, 1);
        }
    }

    // Coalesced b128 stage of box/tgt rows into LDS; weighted-L1 is a global
    // sum, so fold it in right here from registers (weight never touches LDS).
    // Tile is 2816 floats (16B-aligned per block); rows never straddle m.
    constexpr int NV4 = (BLOCK * ROWF) / 4;            // 704 float4s per tile
    const int base4 = blockIdx.x * NV4;
    const int mv4 = m >> 2;                            // uniform scalar bound
    float l1 = 0.f;
#pragma unroll
    for (int j4 = tid; j4 < NV4; j4 += BLOCK) {
        if (j4 < mv4) {
            const float4 b4 = ((const float4*)box)[base4 + j4];
            const float4 t4 = ((const float4*)tgt)[base4 + j4];
            const float4 w4 = ((const float4*)wgt)[base4 + j4];
            ((float4*)sB)[j4] = b4;
            ((float4*)sT)[j4] = t4;
            l1 += fabsf(b4.x - t4.x) * w4.x;
            l1 += fabsf(b4.y - t4.y) * w4.y;
            l1 += fabsf(b4.z - t4.z) * w4.z;
            l1 += fabsf(b4.w - t4.w) * w4.w;
        }
    }
    // scalar tail: only a final partial block with m % 4 != 0 ever runs this
    for (int e = (mv4 << 2) + tid; e < m; e += BLOCK) {
        const float b = box[base + e];
        const float t = tgt[base + e];
        const float w = wgt[base + e];
        sB[e] = b;
        sT[e] = t;
        l1 += fabsf(b - t) * w;
    }
    __syncthreads();

    const int gid = blockIdx.x * BLOCK + tid;
    float gi = 0.f;
    if (gid < n) {
        const float* rb = sB + tid * ROWF;   // stride 11 (odd) -> conflict-free banks
        const float* rt = sT + tid * ROWF;

        // decode BEV: [cx, cy, W=exp, L=exp, yaw]; sin/cos(atan2(s,c)) == s/r, c/r
        const float w1 = __expf(rb[3]);
        const float L1 = __expf(rb[4]);
        const float w2 = __expf(rt[3]);
        const float L2 = __expf(rt[4]);
        const float q1 = rb[6] * rb[6] + rb[7] * rb[7];
        const float q2 = rt[6] * rt[6] + rt[7] * rt[7];
        const float i1 = rsqrtf(q1);
        const float i2 = rsqrtf(q2);
        const float s1 = (q1 > 0.f) ? rb[6] * i1 : 0.f;
        const float c1 = (q1 > 0.f) ? rb[7] * i1 : 1.f;
        const float s2 = (q2 > 0.f) ? rt[6] * i2 : 0.f;
        const float c2 = (q2 > 0.f) ? rt[7] * i2 : 1.f;

        float v1x[4], v1y[4], v2x[4], v2y[4];
        rect_verts(rb[0], rb[1], w1, L1, s1, c1, v1x, v1y);
        rect_verts(rt[0], rt[1], w2, L2, s2, c2, v2x, v2y);

        // clip the subject rect by 4 half-planes; ping-pong buffers A/B,
        // carrying the prev vertex (poly[count-1]) between clips in registers
        float Ax[MAXV], Ay[MAXV], Bx[MAXV], By[MAXV];
        float lastx, lasty;
        int count = clip_rect4(v1x, v1y, v2x[0], v2y[0], v2x[1], v2y[1], Ax, Ay, lastx, lasty);
        count = (count < 3) ? 0 : count;
        count = clip_poly8(Ax, Ay, count, v2x[1], v2y[1], v2x[2], v2y[2], Bx, By, lastx, lasty);
        count = (count < 3) ? 0 : count;
        count = clip_poly8(Bx, By, count, v2x[2], v2y[2], v2x[3], v2y[3], Ax, Ay, lastx, lasty);
        count = (count < 3) ? 0 : count;
        const float acc = clip_area_final(Ax, Ay, count, v2x[3], v2y[3], v2x[0], v2y[0],
                                          lastx, lasty);
        const float inter = fabsf(acc) * 0.5f;

        const float area1 = w1 * L1;
        const float area2 = w2 * L2;
        const float uni = area1 + area2 - inter;

        float xmn = v1x[0], xmx = v1x[0], ymn = v1y[0], ymx = v1y[0];
#pragma unroll
        for (int k = 1; k < 4; ++k) {
            xmn = fminf(xmn, v1x[k]); xmx = fmaxf(xmx, v1x[k]);
            ymn = fminf(ymn, v1y[k]); ymx = fmaxf(ymx, v1y[k]);
        }
#pragma unroll
        for (int k = 0; k < 4; ++k) {
            xmn = fminf(xmn, v2x[k]); xmx = fmaxf(xmx, v2x[k]);
            ymn = fminf(ymn, v2y[k]); ymx = fmaxf(ymx, v2y[k]);
        }
        const float enc = (xmx - xmn) * (ymx - ymn);
        const float iou = inter / (uni + EPSF);
        const float giou = iou - (enc - uni) / (enc + EPSF);
        gi = 1.f - giou;
    }

    // deterministic reduction: wave32 shuffle tree, then serial combine of 8 waves
    const float a = wave_reduce_add(l1);
    const float b = wave_reduce_add(gi);
    const int lane = tid & 31, wid = tid >> 5;
    if (lane == 0) { redA[wid] = a; redB[wid] = b; }
    __syncthreads();
    if (tid == 0) {
        float sa = 0.f, sb = 0.f;
#pragma unroll
        for (int k = 0; k < BLOCK / 32; ++k) { sa += redA[k]; sb += redB[k]; }
        partial[blockIdx.x] = sa;
        partial[gridDim.x + blockIdx.x] = sb;
    }
}

__global__ __launch_bounds__(BLOCK)
void SparseBox3DLoss_finalize_kernel(const float* __restrict__ partial, int nblocks,
                                     const int* __restrict__ avg_factor,
                                     float* __restrict__ out) {
    __shared__ float redA[BLOCK / 32];
    __shared__ float redB[BLOCK / 32];
    const int tid = threadIdx.x;
    float s0 = 0.f, s1 = 0.f;
    for (int i = tid; i < nblocks; i += BLOCK) {   // fixed order -> deterministic
        s0 += partial[i];
        s1 += partial[nblocks + i];
    }
    s0 = wave_reduce_add(s0);
    s1 = wave_reduce_add(s1);
    const int lane = tid & 31, wid = tid >> 5;
    if (lane == 0) { redA[wid] = s0; redB[wid] = s1; }
    __syncthreads();
    if (tid == 0) {
        float sa = 0.f, sb = 0.f;
#pragma unroll
        for (int k = 0; k < BLOCK / 32; ++k) { sa += redA[k]; sb += redB[k]; }
        const float af = fmaxf((float)avg_factor[0], 1.0f);
        out[0] = sa / af;          // loss_box
        out[1] = sb / af;          // loss_giou
    }
}

extern "C" void kernel_launch(void* const* d_in, const int* in_sizes, int n_in,
                              void* d_out, int out_size, void* d_ws, size_t ws_size,
                              hipStream_t stream) {
    const float* box = (const float*)d_in[0];
    const float* tgt = (const float*)d_in[1];
    const float* wgt = (const float*)d_in[2];
    const int*   af  = (const int*)d_in[3];

    const int n = in_sizes[0] / ROWF;
    const int nblocks = (n + BLOCK - 1) / BLOCK;
    float* partial = (float*)d_ws;   // 2 * nblocks floats (16 KB for N=524288)

    SparseBox3DLoss_main_kernel<<<nblocks, BLOCK, 0, stream>>>(box, tgt, wgt, partial, n);
    SparseBox3DLoss_finalize_kernel<<<1, BLOCK, 0, stream>>>(partial, nblocks, af, (float*)d_out);
}